// RelativePositionalEncoding2D_23416161698179
// MI455X (gfx1250) — compile-verified
//
#include <hip/hip_runtime.h>

// RelativePositionalEncoding2D on MI455X (gfx1250, wave32).
//
// out[i,j,e] = W[e, pos(i,j)] + b[e],  pos = clip(idxs[j]-idxs[i], +-32)+32.
// Reference math is onehot(pos) @ W.T + b -- implemented literally on the
// matrix units with exact-fp32 V_WMMA_F32_16X16X4_F32, one-hot A built in
// VALU, B fragments from LDS, bias pre-loaded into the C accumulator.
// Dynamic K-chunk skipping (pos spans <=17 consecutive values per tile)
// keeps it ~1.3 WMMA per 1KB tile, fully hidden under the 512MB store
// stream (store-bandwidth bound: ~22us floor at 23.3 TB/s).
//
// K-chunk bounds are forced into SGPRs via readfirstlane so the WMMA loop
// is a scalar-branch loop: EXEC stays all-1s (ISA requirement for WMMA).

typedef float v2f __attribute__((ext_vector_type(2)));
typedef float v8f __attribute__((ext_vector_type(8)));

namespace {
constexpr int kMaxGap = 32;
constexpr int kN      = 2 * kMaxGap + 1;  // 65
constexpr int kKpad   = 68;               // pad K to multiple of 4 (zeros)
constexpr int kS      = 1024;
constexpr int kE      = 128;
constexpr int kBlk    = 256;              // 8 wave32 -> 8 e-tiles of 16
}

__global__ __launch_bounds__(kBlk)
void relpos2d_wmma_kernel(const int*   __restrict__ idxs,
                          const float* __restrict__ W,     // (E, N) row-major
                          const float* __restrict__ bias,  // (E)
                          float*       __restrict__ out)   // (S, S, E)
{
    // ldsW[e][k] = W[e][k] for k<65, 0 for 65<=k<68  (B-fragment pairs are
    // contiguous in k -> one ds_load_b64 per chunk, conflict-free banks).
    __shared__ float ldsW[kE * kKpad];
    __shared__ float ldsB[kE];

    const int tid = threadIdx.x;

    for (int g = tid; g < kE * kN; g += kBlk) {
        const int e = g / kN;
        const int n = g - e * kN;
        ldsW[e * kKpad + n] = W[g];
    }
    for (int g = tid; g < kE * (kKpad - kN); g += kBlk) {
        const int e = g / (kKpad - kN);
        const int n = kN + (g - e * (kKpad - kN));
        ldsW[e * kKpad + n] = 0.0f;
    }
    if (tid < kE) ldsB[tid] = bias[tid];
    __syncthreads();

    const int lane = tid & 31;
    const int wave = tid >> 5;       // 8 waves -> 8 e-tiles cover E=128
    const int e0   = wave * 16;
    const int n16  = lane & 15;      // row m of A / column N of B,C,D
    const int h    = lane >> 4;      // half selector (K offset 2h)

    const int   i     = blockIdx.y;
    const int   idx_i = idxs[i];
    const float bval  = ldsB[e0 + n16];
    const float* wrow = &ldsW[(e0 + n16) * kKpad];

    const int jb_beg = blockIdx.x * 16;
    for (int jb = jb_beg; jb < jb_beg + 16; ++jb) {
        const int j0    = jb * 16;
        const int idx_j = idxs[j0 + n16];
        int d = idx_j - idx_i;
        d = d < -kMaxGap ? -kMaxGap : (d > kMaxGap ? kMaxGap : d);
        const int pos = d + kMaxGap;          // [0, 64]

        // Wave-wide min/max of pos (halves hold identical values -> 4 steps).
        int pmin = pos, pmax = pos;
        #pragma unroll
        for (int off = 8; off >= 1; off >>= 1) {
            pmin = min(pmin, __shfl_xor(pmin, off, 32));
            pmax = max(pmax, __shfl_xor(pmax, off, 32));
        }
        // Force the (runtime-uniform) chunk bounds into SGPRs: the WMMA loop
        // compiles to a scalar branch and EXEC is never masked around WMMA.
        const int kk_lo = __builtin_amdgcn_readfirstlane(pmin >> 2);
        const int kk_hi = __builtin_amdgcn_readfirstlane(pmax >> 2);  // <= kk_lo+4

        // Seed accumulator with bias: C[m][n] = b[e0+n].
        v8f c;
        #pragma unroll
        for (int v = 0; v < 8; ++v) c[v] = bval;

        for (int kk = kk_lo; kk <= kk_hi; ++kk) {
            const int kbase = kk * 4 + 2 * h;
            v2f b2, a;
            b2.x = wrow[kbase];               // B[k  ][e0+n] = W.T slice
            b2.y = wrow[kbase + 1];           // B[k+1][e0+n]
            a.x  = (pos == kbase)     ? 1.0f : 0.0f;   // one-hot A row
            a.y  = (pos == kbase + 1) ? 1.0f : 0.0f;
            c = __builtin_amdgcn_wmma_f32_16x16x4_f32(
                    /*neg_a=*/false, a, /*neg_b=*/false, b2,
                    /*c_mod=*/(short)0, c, /*reuse_a=*/false, /*reuse_b=*/false);
        }

        // D[m][n]: lane holds rows M = v + 8h, column N = n16.
        const size_t base = ((size_t)i * kS + j0) * kE + e0 + n16;
        #pragma unroll
        for (int v = 0; v < 8; ++v) {
            const int M = v + 8 * h;
            __builtin_nontemporal_store(c[v], &out[base + (size_t)M * kE]);
        }
    }
}

extern "C" void kernel_launch(void* const* d_in, const int* in_sizes, int n_in,
                              void* d_out, int out_size, void* d_ws, size_t ws_size,
                              hipStream_t stream) {
    const int*   idxs = (const int*)d_in[0];    // (1,1024) int32
    const float* W    = (const float*)d_in[1];  // (128,65) fp32
    const float* b    = (const float*)d_in[2];  // (128)    fp32
    float*       out  = (float*)d_out;          // (1,1024,1024,128) fp32

    (void)in_sizes; (void)n_in; (void)out_size; (void)d_ws; (void)ws_size;

    // grid.x: 4 groups of 16 j-blocks (amortizes the 33KB LDS staging so
    // total re-read traffic is ~135MB of L2 hits vs 512MB of stores);
    // grid.y: one block row per i.
    dim3 grid(kS / 16 / 16, kS);
    relpos2d_wmma_kernel<<<grid, dim3(kBlk), 0, stream>>>(idxs, W, b, out);
}